// GCNKGCN_48962627175097
// MI455X (gfx1250) — compile-verified
//
#include <hip/hip_runtime.h>

typedef __attribute__((ext_vector_type(2))) float v2f;
typedef __attribute__((ext_vector_type(8))) float v8f;

#define N_ATOMS 32
#define E_MOL   64
#define IN_CH   64
#define OUT_CH  128
#define HID     256
#define M_MOL   10000
#define E_KG    320000

__device__ __forceinline__ v8f wmma_f32_k4(v2f a, v2f b, v8f c) {
  // D = A(16x4,f32) * B(4x16,f32) + C(16x16,f32), exact f32 WMMA
  return __builtin_amdgcn_wmma_f32_16x16x4_f32(false, a, false, b, (short)0, c,
                                               false, false);
}

// T[32 x N] = A32[32 x K] @ W[K x N].  A32: LDS row-major (stride K).
// W: global row-major (stride N).  Wave-tiled over 2*(N/16) 16x16 output tiles.
// Must be entered by full waves (EXEC all ones); loop bounds are wave-uniform.
__device__ __forceinline__ void gemm32_wmma(const float* __restrict__ A32,
                                            const float* __restrict__ W,
                                            float* __restrict__ T,
                                            int K, int N,
                                            int wave, int nwaves, int lane) {
  const int hi = lane >> 4;   // 0/1: selects K pair (ISA 7.12.2 f32 A layout)
  const int lo = lane & 15;
  const int ntTiles = N >> 4;
  for (int t = wave; t < 2 * ntTiles; t += nwaves) {
    const int mt = t & 1;
    const int nt = t >> 1;
    const int mrow = mt * 16 + lo;
    const int ncol = nt * 16 + lo;
    v8f acc = {};
    for (int kb = 0; kb < K; kb += 4) {
      const int k = kb + 2 * hi;
      v2f a, b;
      a.x = A32[mrow * K + k];        // A lane: row=lo, K=2hi,2hi+1
      a.y = A32[mrow * K + k + 1];
      b.x = W[k * N + ncol];          // B lane: col=lo, K=2hi,2hi+1
      b.y = W[(k + 1) * N + ncol];
      acc = wmma_f32_k4(a, b, acc);
    }
#pragma unroll
    for (int v = 0; v < 8; ++v)       // C/D: vgpr v -> row v+8*hi, col lo
      T[(mt * 16 + v + 8 * hi) * N + nt * 16 + lo] = acc[v];
  }
}

// H[i][c] = dinv[i]^2 * T[i][c] + bias[c] + sum_{e:dst=i} norm[e]*T[src[e]][c]
// (optionally ReLU).  T,H in LDS; edge scatter via LDS float atomics.
__device__ __forceinline__ void aggregate(const float* __restrict__ T,
                                          float* __restrict__ H,
                                          const float* __restrict__ bias,
                                          int Cshift, bool relu,
                                          const int* s_src, const int* s_dst,
                                          const float* s_dinv,
                                          const float* s_norm,
                                          int tid, int nthr) {
  const int Cm1 = (1 << Cshift) - 1;
  for (int idx = tid; idx < (N_ATOMS << Cshift); idx += nthr) {
    const int i = idx >> Cshift;
    const int c = idx & Cm1;
    H[idx] = s_dinv[i] * s_dinv[i] * T[idx] + bias[c];   // self-loop term + bias
  }
  __syncthreads();
  for (int idx = tid; idx < (E_MOL << Cshift); idx += nthr) {
    const int e = idx >> Cshift;
    const int c = idx & Cm1;
    atomicAdd(&H[(s_dst[e] << Cshift) + c],
              s_norm[e] * T[(s_src[e] << Cshift) + c]);
  }
  __syncthreads();
  if (relu) {
    for (int idx = tid; idx < (N_ATOMS << Cshift); idx += nthr)
      H[idx] = fmaxf(H[idx], 0.0f);
  }
  __syncthreads();
}

// One workgroup per molecule: 3 GCN layers fully LDS-resident, then min-pool.
__global__ __launch_bounds__(256) void mol_gnn_kernel(
    const float* __restrict__ mol_x, const int* __restrict__ mei,
    const float* __restrict__ W1, const float* __restrict__ b1,
    const float* __restrict__ W2, const float* __restrict__ b2,
    const float* __restrict__ W3, const float* __restrict__ b3,
    float* __restrict__ embed) {
  __shared__ float Abuf[N_ATOMS * HID];   // 32 KB ping
  __shared__ float Bbuf[N_ATOMS * HID];   // 32 KB pong
  __shared__ float s_dinv[N_ATOMS];
  __shared__ float s_norm[E_MOL];
  __shared__ int   s_deg[N_ATOMS];
  __shared__ int   s_src[E_MOL];
  __shared__ int   s_dst[E_MOL];

  const int mol  = blockIdx.x;
  const int tid  = threadIdx.x;
  const int nthr = blockDim.x;
  const int lane = tid & 31;
  const int wave = tid >> 5;
  const int nw   = nthr >> 5;

  // Load features + edges, build deg/dinv/edge-norm
  const float* X = mol_x + (size_t)mol * (N_ATOMS * IN_CH);
  for (int i = tid; i < N_ATOMS * IN_CH; i += nthr) Abuf[i] = X[i];
  const int* me = mei + (size_t)mol * (2 * E_MOL);
  if (tid < E_MOL) { s_src[tid] = me[tid]; s_dst[tid] = me[E_MOL + tid]; }
  if (tid < N_ATOMS) s_deg[tid] = 1;                    // self loop
  __syncthreads();
  if (tid < E_MOL) atomicAdd(&s_deg[s_dst[tid]], 1);
  __syncthreads();
  if (tid < N_ATOMS) s_dinv[tid] = rsqrtf((float)s_deg[tid]);
  __syncthreads();
  if (tid < E_MOL) s_norm[tid] = s_dinv[s_src[tid]] * s_dinv[s_dst[tid]];
  __syncthreads();

  // Layer 1: [32x64]@[64x256] -> aggregate -> ReLU
  gemm32_wmma(Abuf, W1, Bbuf, IN_CH, HID, wave, nw, lane);
  __syncthreads();
  aggregate(Bbuf, Abuf, b1, 8, true, s_src, s_dst, s_dinv, s_norm, tid, nthr);

  // Layer 2: [32x256]@[256x256] -> aggregate -> ReLU
  gemm32_wmma(Abuf, W2, Bbuf, HID, HID, wave, nw, lane);
  __syncthreads();
  aggregate(Bbuf, Abuf, b2, 8, true, s_src, s_dst, s_dinv, s_norm, tid, nthr);

  // Layer 3: [32x256]@[256x128] -> aggregate (no ReLU)
  gemm32_wmma(Abuf, W3, Bbuf, HID, OUT_CH, wave, nw, lane);
  __syncthreads();
  aggregate(Bbuf, Abuf, b3, 7, false, s_src, s_dst, s_dinv, s_norm, tid, nthr);

  // Min-pool over 32 atoms
  for (int c = tid; c < OUT_CH; c += nthr) {
    float v = Abuf[c];
    for (int r = 1; r < N_ATOMS; ++r) v = fminf(v, Abuf[r * OUT_CH + c]);
    embed[(size_t)mol * OUT_CH + c] = v;
  }
}

// Generic f32 WMMA GEMM: C[M x N] = A[M x K] @ B[K x N], one wave per 16x16 tile.
__global__ __launch_bounds__(256) void wmma_gemm_kernel(
    const float* __restrict__ A, const float* __restrict__ B,
    float* __restrict__ C, int Mrows, int K, int N) {
  const int lane = threadIdx.x & 31;
  const int wave = blockIdx.x * (blockDim.x >> 5) + (threadIdx.x >> 5);
  const int ntT = N >> 4;
  const int nTiles = (Mrows >> 4) * ntT;
  if (wave >= nTiles) return;               // wave-uniform guard
  const int mt = wave / ntT;
  const int nt = wave - mt * ntT;
  const int hi = lane >> 4, lo = lane & 15;
  const int mrow = mt * 16 + lo;
  const int ncol = nt * 16 + lo;
  v8f acc = {};
  for (int kb = 0; kb < K; kb += 4) {
    const int k = kb + 2 * hi;
    v2f a, b;
    a.x = A[(size_t)mrow * K + k];
    a.y = A[(size_t)mrow * K + k + 1];
    b.x = B[(size_t)k * N + ncol];
    b.y = B[(size_t)(k + 1) * N + ncol];
    acc = wmma_f32_k4(a, b, acc);
  }
#pragma unroll
  for (int v = 0; v < 8; ++v)
    C[(size_t)(mt * 16 + v + 8 * hi) * N + nt * 16 + lo] = acc[v];
}

__global__ void kg_deg_init_kernel(float* __restrict__ deg, int n) {
  int i = blockIdx.x * blockDim.x + threadIdx.x;
  if (i < n) deg[i] = 1.0f;                  // self loop
}
__global__ void kg_deg_count_kernel(const int* __restrict__ dst,
                                    float* __restrict__ deg, int ne) {
  int i = blockIdx.x * blockDim.x + threadIdx.x;
  if (i < ne) atomicAdd(&deg[dst[i]], 1.0f);
}
__global__ void kg_dinv_kernel(const float* __restrict__ deg,
                               float* __restrict__ dinv, int n) {
  int i = blockIdx.x * blockDim.x + threadIdx.x;
  if (i < n) dinv[i] = rsqrtf(deg[i]);       // deg >= 1 always
}
__global__ void kg_scale_bias_kernel(const float* __restrict__ XW,
                                     const float* __restrict__ dinv,
                                     const float* __restrict__ bias,
                                     float* __restrict__ H,
                                     int Cshift, int total) {
  int idx = blockIdx.x * blockDim.x + threadIdx.x;
  if (idx >= total) return;
  int i = idx >> Cshift;
  int c = idx & ((1 << Cshift) - 1);
  H[idx] = dinv[i] * dinv[i] * XW[idx] + bias[c];
}
__global__ void kg_edge_kernel(const float* __restrict__ XW,
                               float* __restrict__ H,
                               const float* __restrict__ dinv,
                               const int* __restrict__ src,
                               const int* __restrict__ dst,
                               int Cshift, long long total) {
  long long idx = (long long)blockIdx.x * blockDim.x + threadIdx.x;
  if (idx >= total) return;
  int e = (int)(idx >> Cshift);
  int c = (int)(idx & ((1 << Cshift) - 1));
  int s = src[e], d = dst[e];
  atomicAdd(&H[((size_t)d << Cshift) + c],
            dinv[s] * dinv[d] * XW[((size_t)s << Cshift) + c]);
}
__global__ void relu_kernel(float* __restrict__ h, int n) {
  int i = blockIdx.x * blockDim.x + threadIdx.x;
  if (i < n) h[i] = fmaxf(h[i], 0.0f);
}

extern "C" void kernel_launch(void* const* d_in, const int* in_sizes, int n_in,
                              void* d_out, int out_size, void* d_ws, size_t ws_size,
                              hipStream_t stream) {
  const float* mol_x = (const float*)d_in[0];
  const int*   mei   = (const int*)d_in[1];
  const int*   kei   = (const int*)d_in[2];
  const float* W1 = (const float*)d_in[3];
  const float* b1 = (const float*)d_in[4];
  const float* W2 = (const float*)d_in[5];
  const float* b2 = (const float*)d_in[6];
  const float* W3 = (const float*)d_in[7];
  const float* b3 = (const float*)d_in[8];
  const float* Wk1 = (const float*)d_in[9];
  const float* bk1 = (const float*)d_in[10];
  const float* Wk2 = (const float*)d_in[11];
  const float* bk2 = (const float*)d_in[12];
  float* out = (float*)d_out;

  // Workspace layout (floats): ~30.8 MB total
  float* ws    = (float*)d_ws;
  float* embed = ws;                         // 10000*128
  float* deg   = embed + M_MOL * OUT_CH;     // 10000
  float* dinv  = deg + M_MOL;                // 10000
  float* XW    = dinv + M_MOL;               // 10000*256
  float* h     = XW + (size_t)M_MOL * HID;   // 10000*256
  float* XW2   = h + (size_t)M_MOL * HID;    // 10000*128

  const int* kg_src = kei;
  const int* kg_dst = kei + E_KG;

  // Phase 1: per-molecule GNN + min pool -> embed
  mol_gnn_kernel<<<M_MOL, 256, 0, stream>>>(mol_x, mei, W1, b1, W2, b2, W3, b3,
                                            embed);

  // Phase 2: KG normalization factors
  kg_deg_init_kernel<<<(M_MOL + 255) / 256, 256, 0, stream>>>(deg, M_MOL);
  kg_deg_count_kernel<<<(E_KG + 255) / 256, 256, 0, stream>>>(kg_dst, deg, E_KG);
  kg_dinv_kernel<<<(M_MOL + 255) / 256, 256, 0, stream>>>(deg, dinv, M_MOL);

  // KG layer 1: XW = embed @ Wk1; h = aggregate + bias; ReLU
  {
    int tiles = (M_MOL / 16) * (HID / 16);   // 10000 tiles
    wmma_gemm_kernel<<<(tiles + 7) / 8, 256, 0, stream>>>(embed, Wk1, XW,
                                                          M_MOL, OUT_CH, HID);
  }
  kg_scale_bias_kernel<<<(M_MOL * HID + 255) / 256, 256, 0, stream>>>(
      XW, dinv, bk1, h, 8, M_MOL * HID);
  kg_edge_kernel<<<(int)(((long long)E_KG * HID + 255) / 256), 256, 0, stream>>>(
      XW, h, dinv, kg_src, kg_dst, 8, (long long)E_KG * HID);
  relu_kernel<<<(M_MOL * HID + 255) / 256, 256, 0, stream>>>(h, M_MOL * HID);

  // KG layer 2: XW2 = h @ Wk2; out = aggregate + bias
  {
    int tiles = (M_MOL / 16) * (OUT_CH / 16);  // 5000 tiles
    wmma_gemm_kernel<<<(tiles + 7) / 8, 256, 0, stream>>>(h, Wk2, XW2,
                                                          M_MOL, HID, OUT_CH);
  }
  kg_scale_bias_kernel<<<(M_MOL * OUT_CH + 255) / 256, 256, 0, stream>>>(
      XW2, dinv, bk2, out, 7, M_MOL * OUT_CH);
  kg_edge_kernel<<<(int)(((long long)E_KG * OUT_CH + 255) / 256), 256, 0,
                  stream>>>(XW2, out, dinv, kg_src, kg_dst, 7,
                            (long long)E_KG * OUT_CH);
}